// Warping_77988016161140
// MI455X (gfx1250) — compile-verified
//
#include <hip/hip_runtime.h>

// Warping: out[b,x,y,z] = trilinear(image[b], clamp(grid + ddf))
// Shapes: ddf [4,128,128,128,3] f32, image [4,128,128,128] f32, out [4,128,128,128] f32.
// Memory-bound gather: one thread per output voxel, z fastest.
// NT hints keep the (reused) image resident in the 192MB L2 while streaming ddf/out.

#define FDIM  128
#define FMASK 127

__global__ __launch_bounds__(256) void warp_trilinear_kernel(
    const float* __restrict__ ddf,
    const float* __restrict__ image,
    float* __restrict__ out,
    int n)
{
    int tid = blockIdx.x * blockDim.x + threadIdx.x;
    if (tid >= n) return;

    // decode [b, x, y, z], z fastest (matches row-major [b,f1,f2,f3])
    int z = tid & FMASK;
    int y = (tid >> 7) & FMASK;
    int x = (tid >> 14) & FMASK;
    int b = tid >> 21;

    // --- streamed ddf read (once): non-temporal so it doesn't evict image from L2
    const float* d = ddf + (size_t)tid * 3;
    float dx = __builtin_nontemporal_load(d + 0);
    float dy = __builtin_nontemporal_load(d + 1);
    float dz = __builtin_nontemporal_load(d + 2);

    // warped location, clamped to volume bounds [0, 127]
    float lx = fminf(fmaxf((float)x + dx, 0.0f), 127.0f);
    float ly = fminf(fmaxf((float)y + dy, 0.0f), 127.0f);
    float lz = fminf(fmaxf((float)z + dz, 0.0f), 127.0f);

    float fx = floorf(lx), fy = floorf(ly), fz = floorf(lz);
    int ix0 = (int)fx, iy0 = (int)fy, iz0 = (int)fz;
    int ix1 = min(ix0 + 1, FMASK);
    int iy1 = min(iy0 + 1, FMASK);
    int iz1 = min(iz0 + 1, FMASK);
    float wx = lx - fx, wy = ly - fy, wz = lz - fz;

    // --- 8-corner gather (regular-temporal: image is L2-resident, read ~8x)
    const float* img = image + ((size_t)b << 21);   // b * 128^3
    int rx0 = ix0 << 14, rx1 = ix1 << 14;           // x * 128^2
    int ry0 = iy0 << 7,  ry1 = iy1 << 7;            // y * 128

    float v000 = img[rx0 + ry0 + iz0];
    float v001 = img[rx0 + ry0 + iz1];
    float v010 = img[rx0 + ry1 + iz0];
    float v011 = img[rx0 + ry1 + iz1];
    float v100 = img[rx1 + ry0 + iz0];
    float v101 = img[rx1 + ry0 + iz1];
    float v110 = img[rx1 + ry1 + iz0];
    float v111 = img[rx1 + ry1 + iz1];

    // nested lerps: 7 FMAs
    float c00 = fmaf(wz, v001 - v000, v000);
    float c01 = fmaf(wz, v011 - v010, v010);
    float c10 = fmaf(wz, v101 - v100, v100);
    float c11 = fmaf(wz, v111 - v110, v110);
    float c0  = fmaf(wy, c01 - c00, c00);
    float c1  = fmaf(wy, c11 - c10, c10);
    float res = fmaf(wx, c1 - c0, c0);

    // streamed output: non-temporal store, bypass far-cache dirty retention
    __builtin_nontemporal_store(res, out + tid);
}

extern "C" void kernel_launch(void* const* d_in, const int* in_sizes, int n_in,
                              void* d_out, int out_size, void* d_ws, size_t ws_size,
                              hipStream_t stream) {
    const float* ddf   = (const float*)d_in[0];   // [4,128,128,128,3]
    const float* image = (const float*)d_in[1];   // [4,128,128,128]
    float* out = (float*)d_out;                   // [4,128,128,128]

    int n = out_size;                             // 4 * 128^3 = 8,388,608
    const int block = 256;
    int grid = (n + block - 1) / block;
    warp_trilinear_kernel<<<grid, block, 0, stream>>>(ddf, image, out, n);
}